// DepthCueExtractor_42992622633282
// MI455X (gfx1250) — compile-verified
//
#include <hip/hip_runtime.h>
#include <hip/hip_bf16.h>

// out[b,v,h,f] = sum_w lfi[b,v,h,w] + 128 * h_mask[b,f,h]
// B=8, V=49, H=128, W=128, F=64.  Memory-bound (~39MB @ 23.3TB/s ~= 1.7us).
// Row reduction done on the matrix engine: V_WMMA_F32_16X16X4_F32 with B=ones,
// chained over K (exact f32 math), 16 rows per wave.

typedef __attribute__((ext_vector_type(2))) float v2f;
typedef __attribute__((ext_vector_type(4))) float v4f;
typedef __attribute__((ext_vector_type(8))) float v8f;

#define DCE_B 8
#define DCE_V 49
#define DCE_H 128
#define DCE_W 128
#define DCE_F 64
#define MASK_PITCH 68   // floats per LDS row: 68*4B = 272B, 16B aligned, bank-rotating

__global__ __launch_bounds__(256) void
DepthCueExtractor_42992622633282_kernel(const float* __restrict__ lfi,
                                        const float* __restrict__ h_mask,
                                        float* __restrict__ out) {
    __shared__ float maskT[DCE_H][MASK_PITCH];   // maskT[h][f] = 128 * h_mask[b,f,h]

    const int tid  = threadIdx.x;
    const int wave = tid >> 5;          // 0..7
    const int lane = tid & 31;          // 0..31
    const int l16  = lane & 15;         // column / f-group id
    const int khi  = lane >> 4;         // 0: K={0,1}, 1: K={2,3}

    const int bq = blockIdx.x / DCE_V;  // batch index (v = blockIdx.x % DCE_V, implicit)

    // ---- Phase 1: load h_mask[bq] (F*H = 8192 floats) coalesced, store transposed+scaled ----
    {
        const float* mb = h_mask + (size_t)bq * (DCE_F * DCE_H);
        #pragma unroll
        for (int it = 0; it < (DCE_F * DCE_H) / 256; ++it) {
            int idx = it * 256 + tid;          // idx = f*H + h
            int f = idx >> 7;
            int h = idx & (DCE_H - 1);
            maskT[h][f] = 128.0f * mb[idx];
        }
    }
    __syncthreads();

    // ---- Phase 2: WMMA row-reduction of 16 rows per wave ----
    // Row (= h) handled by this lane's A-matrix slot: m = l16.
    // A f32 16x4 layout: lane L supplies A[m = L&15, K = 2*(L>>4) + {0,1}].
    const size_t blockRow = (size_t)blockIdx.x * DCE_H;     // global row base of this (b,v)
    const int    hBase    = wave * 16;
    const float* aBase    = lfi + (blockRow + hBase + l16) * DCE_W + 2 * khi;

    const v2f ones = {1.0f, 1.0f};
    v8f c0 = {}, c1 = {}, c2 = {}, c3 = {};

#if __has_builtin(__builtin_amdgcn_wmma_f32_16x16x4_f32)
    #pragma unroll
    for (int t = 0; t < 32; t += 4) {
        if ((t & 7) == 0)
            __builtin_prefetch(aBase + 4 * t + 64, 0, 0);   // global_prefetch_b8, ~256B ahead
        v2f a0 = *(const v2f*)(aBase + 4 * (t + 0));
        v2f a1 = *(const v2f*)(aBase + 4 * (t + 1));
        v2f a2 = *(const v2f*)(aBase + 4 * (t + 2));
        v2f a3 = *(const v2f*)(aBase + 4 * (t + 3));
        c0 = __builtin_amdgcn_wmma_f32_16x16x4_f32(false, a0, false, ones, (short)0, c0, false, false);
        c1 = __builtin_amdgcn_wmma_f32_16x16x4_f32(false, a1, false, ones, (short)0, c1, false, false);
        c2 = __builtin_amdgcn_wmma_f32_16x16x4_f32(false, a2, false, ones, (short)0, c2, false, false);
        c3 = __builtin_amdgcn_wmma_f32_16x16x4_f32(false, a3, false, ones, (short)0, c3, false, false);
    }
    v8f c = (c0 + c1) + (c2 + c3);
#else
    // Fallback (no f32 WMMA builtin): shuffle reduction producing the same D layout.
    float partial = 0.0f;
    #pragma unroll
    for (int t = 0; t < 32; ++t) {
        v2f a = *(const v2f*)(aBase + 4 * t);
        partial += a.x + a.y;
    }
    float total = partial + __shfl_xor(partial, 16);   // lane L holds rowsum[L&15]
    v8f c;
    #pragma unroll
    for (int i = 0; i < 8; ++i)
        c[i] = __shfl(total, i + 8 * khi);             // match WMMA D layout
#endif

    // ---- Epilogue ----
    // D layout: c[i] at lane L = rowsum of row (i + 8*(L>>4)); lane l16 covers f = 4*l16..+3.
    #pragma unroll
    for (int i = 0; i < 8; ++i) {
        int hr = hBase + i + 8 * khi;                              // 0..127 within block
        v4f m4 = *(const v4f*)&maskT[hr][4 * l16];                 // ds_load_b128
        float rs = c[i];
        v4f o;
        o.x = rs + m4.x; o.y = rs + m4.y; o.z = rs + m4.z; o.w = rs + m4.w;
        *(v4f*)(out + (blockRow + hr) * DCE_F + 4 * l16) = o;      // global_store_b128
    }
}

extern "C" void kernel_launch(void* const* d_in, const int* in_sizes, int n_in,
                              void* d_out, int out_size, void* d_ws, size_t ws_size,
                              hipStream_t stream) {
    (void)in_sizes; (void)n_in; (void)d_ws; (void)ws_size; (void)out_size;
    const float* lfi    = (const float*)d_in[0];
    // d_in[1] = f_maps: unused by the reference output.
    const float* h_mask = (const float*)d_in[2];
    float* out = (float*)d_out;

    dim3 grid(DCE_B * DCE_V);   // 392 blocks, one per (b, v)
    dim3 block(256);            // 8 waves of 32
    DepthCueExtractor_42992622633282_kernel<<<grid, block, 0, stream>>>(lfi, h_mask, out);
}